// SymbolicPlanner_5806795784850
// MI455X (gfx1250) — compile-verified
//
#include <hip/hip_runtime.h>
#include <cstdint>

#define B_SAMPLES 65536
#define X_DIM 362
#define NCHUNK 512
#define CHUNK 128
#define MAX_TILES 1026   // sum_a ceil(cnt_a/64) <= 1024 + 2

typedef __attribute__((ext_vector_type(2))) float v2f;
typedef __attribute__((ext_vector_type(8))) float v8f;

// packed sizes (float2 elements) per action: nTiles * KQ * 32
#define P1_STRIDE (7 * 2 * 32)      // 448
#define P2_STRIDE (25 * 25 * 32)    // 20000
#define P3_STRIDE (7 * 100 * 32)    // 22400
#define P4_STRIDE (1 * 25 * 32)     // 800

__device__ __forceinline__ v8f wmma4(v2f a, v2f b, v8f c) {
  // D = A(16x4,f32) * B(4x16,f32) + C(16x16,f32)
  return __builtin_amdgcn_wmma_f32_16x16x4_f32(false, a, false, b, (short)0, c,
                                               false, false);
}

// Single-instruction ReLU: for non-NaN IEEE floats, relu(x) is integer max of
// the bit pattern with 0 (positive floats keep their (positive) encoding,
// negative floats -- including -0.0 = INT_MIN -- clamp to +0.0).
// Lowers to one v_max_i32; avoids fmax's canonicalize+max pair.
__device__ __forceinline__ float relu1(float x) {
  return __int_as_float(max(__float_as_int(x), 0));
}

// ---------------------------------------------------------------------------
// Kernel 1: one wave per row — coalesced sentinel scan + neighborhood gather.
// ---------------------------------------------------------------------------
__global__ void k_prep(const float* __restrict__ x, const float* __restrict__ nx,
                       float* __restrict__ attW, float* __restrict__ dltW) {
  int wid = (blockIdx.x * blockDim.x + threadIdx.x) >> 5;
  int lane = threadIdx.x & 31;
  if (wid >= B_SAMPLES) return;
  const float* xr = x + (size_t)wid * X_DIM;
  int best = 0x7FFFFFFF;
#pragma unroll
  for (int j = 0; j < 12; ++j) {
    int idx = lane + j * 32;
    if (idx < X_DIM && xr[idx] == 10.0f) best = min(best, idx);
  }
#pragma unroll
  for (int off = 16; off > 0; off >>= 1) best = min(best, __shfl_xor(best, off, 32));
  int ptr = (best == 0x7FFFFFFF) ? 0 : best;  // argmax of all-False == 0
  if (lane < 8) {
    int cell = 0;
    if (lane == 1) cell = ptr;
    else if (lane == 2) cell = ptr - 19;
    else if (lane == 3) cell = ptr + 19;
    else if (lane == 4) cell = ptr - 1;
    else if (lane == 5) cell = ptr + 1;
    cell = max(0, min(cell, X_DIM - 1));
    float xv = 0.f, dv = 0.f;
    if (lane < 6) {
      xv = xr[cell];
      dv = nx[(size_t)wid * X_DIM + cell] - xv;
    }
    attW[wid * 8 + lane] = xv;
    dltW[wid * 8 + lane] = dv;
  }
}

// ---------------------------------------------------------------------------
// Weight pre-pack into WMMA B-fragment order:
//   dst[a][nt][kq][lane] = (W[k+kb][n], W[k+kb+1][n]); zero-filled padding.
// Makes the GEMM inner loop a single unconditional coalesced b64 load.
// ---------------------------------------------------------------------------
__global__ void k_pack(const float* __restrict__ W, float2* __restrict__ dst,
                       int Kreal, int Nreal, int KQ, int nTiles) {
  int idx = blockIdx.x * blockDim.x + threadIdx.x;
  int total = 3 * nTiles * KQ * 32;
  if (idx >= total) return;
  int lane = idx & 31;
  int t = idx >> 5;
  int kq = t % KQ; t /= KQ;
  int nt = t % nTiles; t /= nTiles;
  int a = t;
  int n = nt * 16 + (lane & 15);
  int k = kq * 4 + ((lane >> 4) << 1);
  const float* Wa = W + (size_t)a * Kreal * Nreal;
  float2 v;
  v.x = (n < Nreal && k < Kreal)       ? Wa[(size_t)k * Nreal + n]       : 0.f;
  v.y = (n < Nreal && (k + 1) < Kreal) ? Wa[(size_t)(k + 1) * Nreal + n] : 0.f;
  dst[idx] = v;
}

// ---------------------------------------------------------------------------
// Deterministic stable partition of sample ids by action (3 buckets).
// ---------------------------------------------------------------------------
__global__ void k_count(const int* __restrict__ act, int* __restrict__ chunkCnt) {
  __shared__ int c[3];
  if (threadIdx.x < 3) c[threadIdx.x] = 0;
  __syncthreads();
  int i = blockIdx.x * CHUNK + threadIdx.x;
  atomicAdd(&c[act[i]], 1);
  __syncthreads();
  if (threadIdx.x < 3) chunkCnt[blockIdx.x * 3 + threadIdx.x] = c[threadIdx.x];
}

__global__ void k_scan(const int* __restrict__ chunkCnt, int* __restrict__ chunkBase,
                       int* __restrict__ cnt) {
  int t = threadIdx.x;
  if (t < 3) {
    int run = 0;
    for (int c = 0; c < NCHUNK; ++c) {
      chunkBase[c * 3 + t] = run;
      run += chunkCnt[c * 3 + t];
    }
    cnt[t] = run;
  }
}

__global__ void k_scatter(const int* __restrict__ act, const int* __restrict__ chunkBase,
                          const int* __restrict__ cnt, int* __restrict__ perm) {
  __shared__ int aL[CHUNK];
  int t = threadIdx.x;
  int i = blockIdx.x * CHUNK + t;
  int a = act[i];
  aL[t] = a;
  __syncthreads();
  int rank = 0;
  for (int j = 0; j < t; ++j) rank += (aL[j] == a) ? 1 : 0;
  int seg = (a > 0 ? cnt[0] : 0) + (a > 1 ? cnt[1] : 0);
  perm[seg + chunkBase[blockIdx.x * 3 + a] + rank] = i;
}

// ---------------------------------------------------------------------------
// One fused MLP layer over a 64-sample tile (compile-time geometry):
//   hout[64][.] = relu(hin[64][4*KQ] @ W + bias)
// Inner loop: 1 coalesced b64 weight load + 4 ds_load_2addr A-frags + 4 WMMAs.
// k-loop unrolled x4 so consecutive k-steps fold into ds immediate offsets.
// ---------------------------------------------------------------------------
template <int KQ, int NTILES, int SIN, int SOUT>
__device__ __forceinline__ void mlp_layer(
    const v2f* __restrict__ Wp,        // packed, this action: [NTILES][KQ][32]
    const float* __restrict__ bias, int Nreal,
    const float* __restrict__ hin, float* __restrict__ hout,
    int wave, int lane) {
  int nlo = lane & 15;
  int kb0 = (lane >> 4) << 1;           // lanes 0-15: K+0/1, lanes 16-31: K+2/3
  int rowAdd = (lane & 16) ? 8 : 0;     // C/D: VGPR v -> row v (lo) / v+8 (hi)
  for (int nt = wave; nt < NTILES; nt += 8) {
    int nG = nt * 16 + nlo;
    float bv = (nG < Nreal) ? bias[nG] : 0.f;  // padded cols: 0 bias + 0 weights
    v8f acc0, acc1, acc2, acc3;
#pragma unroll
    for (int v = 0; v < 8; ++v) { acc0[v] = bv; acc1[v] = bv; acc2[v] = bv; acc3[v] = bv; }
    const v2f* wp = Wp + (size_t)nt * (KQ * 32) + lane;
#pragma unroll 4
    for (int kq = 0; kq < KQ; ++kq) {
      v2f bf = wp[kq * 32];             // one global_load_b64, fully coalesced
      int k = kq * 4 + kb0;
      v2f a0, a1, a2, a3;
      a0.x = hin[(nlo)*SIN + k];      a0.y = hin[(nlo)*SIN + k + 1];
      a1.x = hin[(16 + nlo)*SIN + k]; a1.y = hin[(16 + nlo)*SIN + k + 1];
      a2.x = hin[(32 + nlo)*SIN + k]; a2.y = hin[(32 + nlo)*SIN + k + 1];
      a3.x = hin[(48 + nlo)*SIN + k]; a3.y = hin[(48 + nlo)*SIN + k + 1];
      acc0 = wmma4(a0, bf, acc0);
      acc1 = wmma4(a1, bf, acc1);
      acc2 = wmma4(a2, bf, acc2);
      acc3 = wmma4(a3, bf, acc3);
    }
#pragma unroll
    for (int v = 0; v < 8; ++v) {       // unconditional: padded cols are 0
      hout[(v + rowAdd) * SOUT + nG]      = relu1(acc0[v]);
      hout[(16 + v + rowAdd) * SOUT + nG] = relu1(acc1[v]);
      hout[(32 + v + rowAdd) * SOUT + nG] = relu1(acc2[v]);
      hout[(48 + v + rowAdd) * SOUT + nG] = relu1(acc3[v]);
    }
  }
}

// ---------------------------------------------------------------------------
// Kernel: 64-sample tile of a single expert, full MLP + squared-error sum.
// ---------------------------------------------------------------------------
__global__ __launch_bounds__(256) void k_mlp(
    const float* __restrict__ attW, const float* __restrict__ dltW,
    const int* __restrict__ perm, const int* __restrict__ cnt,
    const v2f* __restrict__ pW1, const float* __restrict__ b1,
    const v2f* __restrict__ pW2, const float* __restrict__ b2,
    const v2f* __restrict__ pW3, const float* __restrict__ b3,
    const v2f* __restrict__ pW4, const float* __restrict__ b4,
    float* __restrict__ partials) {
  __shared__ float attL[64 * 9];     // odd strides => conflict-free A-frag reads
  __shared__ float dltL[64 * 8];
  __shared__ float h13[64 * 113];    // h1, then reused for h3
  __shared__ float h2s[64 * 401];
  __shared__ float wsum[4];

  int g = blockIdx.x;
  int c0 = cnt[0], c1 = cnt[1], c2 = cnt[2];
  int t0 = (c0 + 63) >> 6, t1 = (c1 + 63) >> 6, t2 = (c2 + 63) >> 6;
  int a, tile, segS, cA;
  if (g < t0)                { a = 0; tile = g;           segS = 0;       cA = c0; }
  else if (g < t0 + t1)      { a = 1; tile = g - t0;      segS = c0;      cA = c1; }
  else if (g < t0 + t1 + t2) { a = 2; tile = g - t0 - t1; segS = c0 + c1; cA = c2; }
  else { if (threadIdx.x == 0) partials[g] = 0.f; return; }
  int base = tile << 6;
  int tid = threadIdx.x;

  for (int idx = tid; idx < 64 * 9; idx += 256) {
    int r = idx / 9, c = idx - r * 9;
    float v = 0.f;
    if (c < 6 && base + r < cA) v = attW[(size_t)perm[segS + base + r] * 8 + c];
    attL[idx] = v;
  }
  for (int idx = tid; idx < 64 * 8; idx += 256) {
    int r = idx >> 3, c = idx & 7;
    float v = 0.f;
    if (c < 6 && base + r < cA) v = dltW[(size_t)perm[segS + base + r] * 8 + c];
    dltL[idx] = v;
  }
  __syncthreads();

  int wave = tid >> 5, lane = tid & 31;
  const v2f* W1a = pW1 + (size_t)a * P1_STRIDE; const float* b1a = b1 + a * 100;
  const v2f* W2a = pW2 + (size_t)a * P2_STRIDE; const float* b2a = b2 + a * 400;
  const v2f* W3a = pW3 + (size_t)a * P3_STRIDE; const float* b3a = b3 + a * 100;
  const v2f* W4a = pW4 + (size_t)a * P4_STRIDE; const float* b4a = b4 + a * 6;

  mlp_layer<2, 7, 9, 113>(W1a, b1a, 100, attL, h13, wave, lane);
  __syncthreads();
  mlp_layer<25, 25, 113, 401>(W2a, b2a, 400, h13, h2s, wave, lane);
  __syncthreads();
  mlp_layer<100, 7, 401, 113>(W3a, b3a, 100, h2s, h13, wave, lane);
  __syncthreads();

  // Layer 4 (100 -> 6, packed/zero-padded) fused with error accumulation.
  if (wave < 4) {
    int mt = wave;
    int nlo = lane & 15;
    int kb0 = (lane >> 4) << 1;
    int rowAdd = (lane & 16) ? 8 : 0;
    float bv = (nlo < 6) ? b4a[nlo] : 0.f;
    v8f acc;
#pragma unroll
    for (int v = 0; v < 8; ++v) acc[v] = bv;
    const v2f* wp = W4a + lane;
#pragma unroll 5
    for (int kq = 0; kq < 25; ++kq) {
      v2f bf = wp[kq * 32];
      int k = kq * 4 + kb0;
      int m = mt * 16 + nlo;
      v2f af;
      af.x = h13[m * 113 + k];
      af.y = h13[m * 113 + k + 1];
      acc = wmma4(af, bf, acc);
    }
    float s = 0.f;
#pragma unroll
    for (int v = 0; v < 8; ++v) {
      int row = mt * 16 + v + rowAdd;
      if (nlo < 6 && base + row < cA) {
        float e = acc[v] - dltL[row * 8 + nlo];
        s += e * e;
      }
    }
#pragma unroll
    for (int off = 16; off > 0; off >>= 1) s += __shfl_xor(s, off, 32);
    if (lane == 0) wsum[mt] = s;
  }
  __syncthreads();
  if (tid == 0) partials[g] = wsum[0] + wsum[1] + wsum[2] + wsum[3];
}

// Fixed-tree reduction => deterministic scalar.
__global__ void k_reduce(const float* __restrict__ partials, float* __restrict__ out) {
  __shared__ float s[512];
  int t = threadIdx.x;
  float v = 0.f;
  if (t < MAX_TILES) v = partials[t];
  if (t + 512 < MAX_TILES) v += partials[t + 512];
  s[t] = v;
  __syncthreads();
  for (int off = 256; off > 0; off >>= 1) {
    if (t < off) s[t] += s[t + off];
    __syncthreads();
  }
  if (t == 0) out[0] = s[0] / (float)(B_SAMPLES * 6);
}

extern "C" void kernel_launch(void* const* d_in, const int* in_sizes, int n_in,
                              void* d_out, int out_size, void* d_ws, size_t ws_size,
                              hipStream_t stream) {
  (void)in_sizes; (void)n_in; (void)out_size; (void)ws_size;
  const float* x  = (const float*)d_in[0];
  const float* nx = (const float*)d_in[1];
  const int*   ac = (const int*)d_in[2];
  const float* W1 = (const float*)d_in[3];
  const float* b1 = (const float*)d_in[4];
  const float* W2 = (const float*)d_in[5];
  const float* b2 = (const float*)d_in[6];
  const float* W3 = (const float*)d_in[7];
  const float* b3 = (const float*)d_in[8];
  const float* W4 = (const float*)d_in[9];
  const float* b4 = (const float*)d_in[10];
  float* out = (float*)d_out;

  float* attW      = (float*)d_ws;                  // B*8 floats
  float* dltW      = attW + B_SAMPLES * 8;          // B*8 floats
  int*   perm      = (int*)(dltW + B_SAMPLES * 8);  // B ints
  int*   chunkCnt  = perm + B_SAMPLES;              // NCHUNK*3
  int*   chunkBase = chunkCnt + NCHUNK * 3;         // NCHUNK*3
  int*   cnt       = chunkBase + NCHUNK * 3;        // 3 (+1 pad)
  float* partials  = (float*)(cnt + 4);             // MAX_TILES (+pad to even)
  float2* pW1 = (float2*)(partials + MAX_TILES + 2);        // 8B aligned
  float2* pW2 = pW1 + 3 * P1_STRIDE;
  float2* pW3 = pW2 + 3 * P2_STRIDE;
  float2* pW4 = pW3 + 3 * P3_STRIDE;

  k_prep<<<B_SAMPLES / 8, 256, 0, stream>>>(x, nx, attW, dltW);
  k_pack<<<(3 * P1_STRIDE + 255) / 256, 256, 0, stream>>>(W1, pW1, 6,   100, 2,   7);
  k_pack<<<(3 * P2_STRIDE + 255) / 256, 256, 0, stream>>>(W2, pW2, 100, 400, 25,  25);
  k_pack<<<(3 * P3_STRIDE + 255) / 256, 256, 0, stream>>>(W3, pW3, 400, 100, 100, 7);
  k_pack<<<(3 * P4_STRIDE + 255) / 256, 256, 0, stream>>>(W4, pW4, 100, 6,   25,  1);
  k_count<<<NCHUNK, CHUNK, 0, stream>>>(ac, chunkCnt);
  k_scan<<<1, 64, 0, stream>>>(chunkCnt, chunkBase, cnt);
  k_scatter<<<NCHUNK, CHUNK, 0, stream>>>(ac, chunkBase, cnt, perm);
  k_mlp<<<MAX_TILES, 256, 0, stream>>>(attW, dltW, perm, cnt,
                                       (const v2f*)pW1, b1, (const v2f*)pW2, b2,
                                       (const v2f*)pW3, b3, (const v2f*)pW4, b4,
                                       partials);
  k_reduce<<<1, 512, 0, stream>>>(partials, out);
}